// LogEig_16063177687357
// MI455X (gfx1250) — compile-verified
//
#include <hip/hip_runtime.h>
#include <math.h>

typedef __attribute__((ext_vector_type(2))) float v2f;
typedef __attribute__((ext_vector_type(4))) float v4f;
typedef __attribute__((ext_vector_type(8))) float v8f;

#define LDW   68            // LDS row stride (floats): spreads banks, keeps 16B align
#define MATSZ (64 * LDW)

// C = p0 * (A @ B) + p1 * I   for 64x64 row-major matrices in LDS.
// 128 threads = 4 waves; wave w computes rows [16w, 16w+16), i.e. 4 C-tiles of 16x16.
// Fragment layouts per CDNA5 ISA 7.12.2 (V_WMMA_F32_16X16X4_F32):
//   A 16x4:  lanes 0-15 -> M=lane, VGPR0/1 = K0/K1; lanes 16-31 -> K2/K3
//   B 4x16:  VGPR r -> K=r (lanes 0-15) and K=r+2 (lanes 16-31), N = lane&15
//   C 16x16: VGPR r -> M=r (lanes 0-15) / M=r+8 (lanes 16-31), N = lane&15
__device__ __forceinline__ void mm64(const float* __restrict__ A,
                                     const float* __restrict__ B,
                                     float* __restrict__ C,
                                     float p0, float p1)
{
  const int tid  = threadIdx.x;
  const int lane = tid & 31;
  const int half = lane >> 4;      // 0 or 1
  const int l15  = lane & 15;
  const int m0   = (tid >> 5) << 4;  // this wave's row block
  const float* arow = A + (m0 + l15) * LDW;

  v8f acc[4] = {};
  #pragma unroll
  for (int k = 0; k < 64; k += 4) {
    v2f a;
    a.x = arow[k + 2 * half + 0];
    a.y = arow[k + 2 * half + 1];
    const float* b0 = B + (k + 2 * half) * LDW + l15;  // K = k+2*half
    const float* b1 = b0 + LDW;                        // K = k+2*half+1
    #pragma unroll
    for (int nt = 0; nt < 4; ++nt) {
      v2f b;
      b.x = b0[nt * 16];
      b.y = b1[nt * 16];
      acc[nt] = __builtin_amdgcn_wmma_f32_16x16x4_f32(
          false, a, false, b, (short)0, acc[nt], false, false);
    }
  }

  const int rbase = m0 + half * 8;
  #pragma unroll
  for (int nt = 0; nt < 4; ++nt) {
    const int col = nt * 16 + l15;
    #pragma unroll
    for (int r = 0; r < 8; ++r) {
      const int row = rbase + r;
      float v = p0 * acc[nt][r];
      if (row == col) v += p1;
      C[row * LDW + col] = v;
    }
  }
  __syncthreads();   // C visible to all waves; also closes WAR on A/B for next mm
}

// P = a * I (zero elsewhere), 128 threads, vectorized
__device__ __forceinline__ void setIdent(float* P, float a)
{
  const int tid = threadIdx.x;
  #pragma unroll
  for (int j = 0; j < 8; ++j) {
    const int li  = (tid + 128 * j) * 4;
    const int row = li >> 6, col = li & 63;
    v4f v = {0.f, 0.f, 0.f, 0.f};
    const int d = row - col;
    if (d >= 0 && d < 4) v[d] = a;
    *(v4f*)(P + row * LDW + col) = v;
  }
  __syncthreads();
}

__global__ void __launch_bounds__(128)
logm_spd64_kernel(const float* __restrict__ X, float* __restrict__ Out)
{
  extern __shared__ float smem[];
  float* b0  = smem;
  float* b1  = smem + MATSZ;
  float* b2  = smem + 2 * MATSZ;
  float* b3  = smem + 3 * MATSZ;
  float* red = smem + 4 * MATSZ;   // [128]
  float* sc  = red + 128;          // [2]: 1/c, log(c)

  const int    tid  = threadIdx.x;
  const size_t base = (size_t)blockIdx.x * 4096;

  // ---- load X into LDS (coalesced float4) + Frobenius norm reduction ----
  float s = 0.f;
  #pragma unroll
  for (int j = 0; j < 8; ++j) {
    const int li  = (tid + 128 * j) * 4;
    const int row = li >> 6, col = li & 63;
    v4f v = *(const v4f*)(X + base + li);
    *(v4f*)(b0 + row * LDW + col) = v;
    s += v.x * v.x + v.y * v.y + v.z * v.z + v.w * v.w;
  }
  red[tid] = s;
  __syncthreads();
  for (int off = 64; off > 0; off >>= 1) {
    if (tid < off) red[tid] += red[tid + off];
    __syncthreads();
  }
  if (tid == 0) {
    const float c = sqrtf(red[0]) + 1e-30f;
    sc[0] = 1.0f / c;
    sc[1] = logf(c);
  }
  __syncthreads();
  const float inv  = sc[0];
  const float logc = sc[1];

  // ---- A = X / c (eigenvalues now in (0,1]) ----
  #pragma unroll
  for (int j = 0; j < 8; ++j) {
    const int li  = (tid + 128 * j) * 4;
    const int row = li >> 6, col = li & 63;
    v4f v = *(v4f*)(b0 + row * LDW + col);
    v.x *= inv; v.y *= inv; v.z *= inv; v.w *= inv;
    *(v4f*)(b0 + row * LDW + col) = v;
  }
  __syncthreads();

  float *pY = b0, *pZ = b1, *pT = b2, *pW = b3;

  // ---- L = 5 repeated square roots via coupled Newton-Schulz ----
  for (int lvl = 0; lvl < 5; ++lvl) {
    setIdent(pZ, 1.0f);                     // Z = I
    const int ni = (lvl == 0) ? 12 : 9;
    for (int it = 0; it < ni; ++it) {
      mm64(pZ, pY, pT, -0.5f, 1.5f);        // T = 1.5 I - 0.5 Z Y
      mm64(pY, pT, pW, 1.0f, 0.0f);         // Y' = Y T
      mm64(pT, pZ, pY, 1.0f, 0.0f);         // Z' = T Z   (old Y buf is dead)
      float* t = pY;                        // rotate buffers
      pY = pW;                              // Y'
      pW = pZ;                              // free
      pZ = t;                               // Z'
    }
  }

  // ---- E = A^{1/32} - I ----
  if (tid < 64) pY[tid * LDW + tid] -= 1.0f;
  __syncthreads();

  // ---- log(I+E) ~ E*(a1 I + E(a2 I + ... )) , a_k = (-1)^{k+1}/k, 7 terms ----
  setIdent(pZ, 1.0f / 7.0f);                // P = a7 I
  #pragma unroll
  for (int k = 0; k < 6; ++k) {
    const float coef[6] = {-1.f / 6.f, 1.f / 5.f, -1.f / 4.f,
                            1.f / 3.f, -1.f / 2.f, 1.f};
    mm64(pY, pZ, pT, 1.0f, coef[k]);        // P = a_k I + E P
    float* t = pZ; pZ = pT; pT = t;
  }

  // ---- Out = 2^5 * E P + log(c) * I ----
  mm64(pY, pZ, pW, 32.0f, logc);

  #pragma unroll
  for (int j = 0; j < 8; ++j) {
    const int li  = (tid + 128 * j) * 4;
    const int row = li >> 6, col = li & 63;
    *(v4f*)(Out + base + li) = *(const v4f*)(pW + row * LDW + col);
  }
}

extern "C" void kernel_launch(void* const* d_in, const int* in_sizes, int n_in,
                              void* d_out, int out_size, void* d_ws, size_t ws_size,
                              hipStream_t stream) {
  const float* x   = (const float*)d_in[0];
  float*       out = (float*)d_out;
  const int nmat   = in_sizes[0] / 4096;          // 8192 matrices of 64x64
  const size_t shmem = (size_t)(4 * MATSZ + 128 + 2) * sizeof(float);  // ~70 KB

  static_assert(4 * MATSZ * sizeof(float) < 320 * 1024, "fits WGP LDS");
  hipFuncSetAttribute((const void*)logm_spd64_kernel,
                      hipFuncAttributeMaxDynamicSharedMemorySize, (int)shmem);
  logm_spd64_kernel<<<dim3(nmat), dim3(128), shmem, stream>>>(x, out);
}